// CRF_43989055045684
// MI455X (gfx1250) — compile-verified
//
#include <hip/hip_runtime.h>
#include <stdint.h>

#define TAGS   50
#define TP     52      // padded tag count (13 float4 chunks)
#define NCH    13
#define SEQ    2048
#define BAT    256
#define NEGV   (-10000.0f)
#define STARTT 48
#define ENDT   49
#define CCHUNK 32      // backtrack chunks
#define CHLEN  64      // steps per chunk over s in [1, SEQ-1]
#define NBUF   4       // emission ring buffers (3 async loads in flight)

#if defined(__HIP_DEVICE_COMPILE__)
#if __has_builtin(__builtin_amdgcn_wave_barrier)
#define WAVE_BARRIER() __builtin_amdgcn_wave_barrier()
#else
#define WAVE_BARRIER() do {} while (0)
#endif
#else
#define WAVE_BARRIER() do {} while (0)
#endif

__device__ __forceinline__ uint32_t lds_addr_of(const void* p) {
  return (uint32_t)(uintptr_t)p;  // addrspace(3) ptr -> LDS byte offset
}

// Async copy 8 bytes per lane from the global address saddr plus voff to the LDS address in vdst; non-temporal
// (feats is a one-shot stream; keep the L2 for the 26MB backtrack table).
__device__ __forceinline__ void async_ld_b64_nt(void* lds_dst, const void* gbase, uint32_t voff) {
  asm volatile("global_load_async_to_lds_b64 %0, %1, %2 th:TH_LOAD_NT"
               :
               : "v"(lds_addr_of(lds_dst)), "v"(voff),
                 "s"((uint64_t)(uintptr_t)gbase)
               : "memory");
}
__device__ __forceinline__ void wait_async_le3() {
  asm volatile("s_wait_asynccnt 0x3" ::: "memory");
}
__device__ __forceinline__ void wait_async0() {
  asm volatile("s_wait_asynccnt 0x0" ::: "memory");
}

// ---------------- forward Viterbi scan: one wave32 per batch row ----------------
__global__ __launch_bounds__(32) void crf_forward(
    const float* __restrict__ feats, const float* __restrict__ masks,
    const float* __restrict__ trans, float* __restrict__ out_scores,
    uint8_t* __restrict__ pre, int* __restrict__ next_tag) {
  __shared__ float ltr[TAGS * TP];     // transitions, row = next tag, padded cols = NEG
  __shared__ float lfs[TP];            // running scores (padded with NEG)
  __shared__ float lfeat[NBUF][TP];    // emission ring (async-to-LDS target)

  const int lane = threadIdx.x;
  const int b = blockIdx.x;

  for (int r = 0; r < TAGS; ++r)
    for (int c = lane; c < TP; c += 32)
      ltr[r * TP + c] = (c < TAGS) ? trans[r * TAGS + c] : NEGV;
  __syncthreads();

  const int t0 = 2 * lane, t1 = t0 + 1;
  const bool act = (t0 < TAGS);  // lanes 0..24 own two tags each

  // Keep both transition rows in VGPRs (104 regs) -> per-step LDS traffic is fs only.
  float4 rowA[NCH], rowB[NCH];
  {
    int rA = act ? t0 : 0, rB = act ? t1 : 0;
#pragma unroll
    for (int c = 0; c < NCH; ++c) {
      rowA[c] = *reinterpret_cast<const float4*>(&ltr[rA * TP + 4 * c]);
      rowB[c] = *reinterpret_cast<const float4*>(&ltr[rB * TP + 4 * c]);
    }
  }

  if (lane < 26) {  // init fs[0..51]
    float2 v;
    v.x = (t0 == STARTT) ? 0.0f : NEGV;
    v.y = NEGV;  // STARTT is even, never in .y
    *reinterpret_cast<float2*>(&lfs[t0]) = v;
  }
  WAVE_BARRIER();

  const float* fb = feats + (size_t)b * SEQ * TAGS;
  const uint32_t voff = (uint32_t)(lane * 8);

  // one Viterbi step: emissions for step s are in lfeat[buf]
  auto step = [&](int s, int buf) {
    float best0 = -3.0e38f, best1 = -3.0e38f;
    int a0 = 0, a1 = 0;
#pragma unroll
    for (int c = 0; c < NCH; ++c) {
      float4 f = *reinterpret_cast<const float4*>(&lfs[4 * c]);  // broadcast
#define UPD(FE, TA, TB, J)                                                  \
      {                                                                     \
        float c0 = (FE) + (TA);                                             \
        bool g0 = c0 > best0; best0 = g0 ? c0 : best0; a0 = g0 ? (J) : a0;  \
        float c1 = (FE) + (TB);                                             \
        bool g1 = c1 > best1; best1 = g1 ? c1 : best1; a1 = g1 ? (J) : a1;  \
      }
      UPD(f.x, rowA[c].x, rowB[c].x, 4 * c + 0)
      UPD(f.y, rowA[c].y, rowB[c].y, 4 * c + 1)
      UPD(f.z, rowA[c].z, rowB[c].z, 4 * c + 2)
      UPD(f.w, rowA[c].w, rowB[c].w, 4 * c + 3)
#undef UPD
    }
    float m = masks[(size_t)b * SEQ + s];  // wave-uniform -> SMEM
    float2 em = {0.0f, 0.0f};
    float2 fold = {NEGV, NEGV};
    if (act) {
      em = *reinterpret_cast<const float2*>(&lfeat[buf][t0]);
      fold = *reinterpret_cast<const float2*>(&lfs[t0]);
    }
    float w0 = (best0 + em.x) * m + fold.x * (1.0f - m);
    float w1 = (best1 + em.y) * m + fold.y * (1.0f - m);
    WAVE_BARRIER();
    if (act) {
      float2 nv; nv.x = w0; nv.y = w1;
      *reinterpret_cast<float2*>(&lfs[t0]) = nv;
      uint16_t packed = (uint16_t)((a0 & 0xff) | ((a1 & 0xff) << 8));
      *reinterpret_cast<uint16_t*>(&pre[((size_t)s * BAT + b) * TAGS + t0]) = packed;
    }
    WAVE_BARRIER();
  };

  // prime the ring: 3 loads in flight
  if (lane < 25) {
#pragma unroll
    for (int p = 0; p < NBUF - 1; ++p)
      async_ld_b64_nt(&lfeat[p][2 * lane], fb + (size_t)p * TAGS, voff);
  }

  // main loop: issue s+3, wait ASYNCcnt<=3 (in-order completion => buf for s ready)
  for (int s = 0; s < SEQ - (NBUF - 1); ++s) {
    if (lane < 25)
      async_ld_b64_nt(&lfeat[(s + NBUF - 1) & (NBUF - 1)][2 * lane],
                      fb + (size_t)(s + NBUF - 1) * TAGS, voff);
    wait_async_le3();
    step(s, s & (NBUF - 1));
  }
  // tail: everything already issued; drain and finish
  wait_async0();
  for (int s = SEQ - (NBUF - 1); s < SEQ; ++s) step(s, s & (NBUF - 1));

  // final: fs += trans[ENDT,:]; wave-reduce max/argmax (first-index tie-break)
  float te0 = act ? ltr[ENDT * TP + t0] : NEGV;
  float te1 = act ? ltr[ENDT * TP + t1] : NEGV;
  float2 ff = {NEGV, NEGV};
  if (act) ff = *reinterpret_cast<const float2*>(&lfs[t0]);
  float v0 = ff.x + te0;
  int a0 = t0;
  float v1 = ff.y + te1;
  if (act && v1 > v0) { v0 = v1; a0 = t1; }
  if (!act) { v0 = -3.0e38f; a0 = 0x7fffffff; }
  for (int off = 16; off > 0; off >>= 1) {
    float ov = __shfl_xor(v0, off, 32);
    int oa = __shfl_xor(a0, off, 32);
    if (ov > v0 || (ov == v0 && oa < a0)) { v0 = ov; a0 = oa; }
  }
  if (lane == 0) { out_scores[b] = v0; next_tag[b] = a0; }
}

// ------------- backtrack phase 1: compose tag->tag maps per chunk -------------
__global__ __launch_bounds__(64) void crf_compose(
    const float* __restrict__ masks, const uint8_t* __restrict__ pre,
    uint8_t* __restrict__ M) {
  int b = blockIdx.x / CCHUNK;
  int k = blockIdx.x % CCHUNK;
  int t = threadIdx.x;
  if (t >= TAGS) return;
  int lo = k * CHLEN + 1;
  int hi = lo + CHLEN - 1; if (hi > SEQ - 1) hi = SEQ - 1;
  int cur = t;
  for (int s = hi; s >= lo; --s) {
    float m = masks[(size_t)b * SEQ + s];
    int g = pre[((size_t)s * BAT + b) * TAGS + cur];
    cur = (m != 0.0f) ? g : cur;
  }
  M[((size_t)b * CCHUNK + k) * TAGS + t] = (uint8_t)cur;
}

// ------------- phase 2: chain chunk maps from next_tag (short serial) -------------
__global__ __launch_bounds__(256) void crf_select(
    const int* __restrict__ next_tag, const uint8_t* __restrict__ M,
    int* __restrict__ entry, float* __restrict__ out) {
  int b = threadIdx.x;
  int t = next_tag[b];
  for (int k = CCHUNK - 1; k >= 0; --k) {
    entry[b * CCHUNK + k] = t;              // tag entering chunk k's highest s
    t = M[((size_t)b * CCHUNK + k) * TAGS + t];
  }
  out[BAT + (size_t)b * SEQ + 0] = (float)t;  // seq[b][0]
}

// ------------- phase 3: re-walk each chunk in parallel, emit tags -------------
__global__ __launch_bounds__(256) void crf_emit(
    const float* __restrict__ masks, const uint8_t* __restrict__ pre,
    const int* __restrict__ entry, float* __restrict__ out) {
  int idx = blockIdx.x * blockDim.x + threadIdx.x;
  if (idx >= BAT * CCHUNK) return;
  int b = idx / CCHUNK, k = idx % CCHUNK;
  int lo = k * CHLEN + 1;
  int hi = lo + CHLEN - 1; if (hi > SEQ - 1) hi = SEQ - 1;
  int t = entry[b * CCHUNK + k];
  for (int s = hi; s >= lo; --s) {
    float m = masks[(size_t)b * SEQ + s];
    bool mi = (m != 0.0f);
    out[BAT + (size_t)b * SEQ + s] = (float)(mi ? t : ENDT);
    if (mi) t = pre[((size_t)s * BAT + b) * TAGS + t];
  }
}

extern "C" void kernel_launch(void* const* d_in, const int* in_sizes, int n_in,
                              void* d_out, int out_size, void* d_ws, size_t ws_size,
                              hipStream_t stream) {
  (void)in_sizes; (void)n_in; (void)out_size; (void)ws_size;
  const float* feats = (const float*)d_in[0];
  const float* masks = (const float*)d_in[1];
  const float* trans = (const float*)d_in[2];
  float* out = (float*)d_out;

  uint8_t* pre = (uint8_t*)d_ws;                     // SEQ*BAT*TAGS = 26,214,400 B
  size_t preB = (size_t)SEQ * BAT * TAGS;
  int* nt = (int*)(pre + preB);                      // BAT ints
  uint8_t* M = (uint8_t*)(nt + BAT);                 // BAT*CCHUNK*TAGS = 409,600 B
  int* entry = (int*)(M + (size_t)BAT * CCHUNK * TAGS);  // BAT*CCHUNK ints

  crf_forward<<<BAT, 32, 0, stream>>>(feats, masks, trans, out, pre, nt);
  crf_compose<<<BAT * CCHUNK, 64, 0, stream>>>(masks, pre, M);
  crf_select<<<1, BAT, 0, stream>>>(nt, M, entry, out);
  crf_emit<<<(BAT * CCHUNK + 255) / 256, 256, 0, stream>>>(masks, pre, entry, out);
}